// KalmanFilter_59648505807192
// MI455X (gfx1250) — compile-verified
//
#include <hip/hip_runtime.h>
#include <math.h>

// Kalman filter scan: T=8192 sequential steps, S=128 state, O=64 obs.
// Single persistent 1024-thread workgroup (32 waves) on one WGP.
// All per-step state in LDS (~314KB of the 320KB WGP pool).
// fp32 matmuls via V_WMMA_F32_16X16X4_F32, K-loop software-pipelined
// so LDS fragment loads overlap the WMMA issue (avoid s_wait_dscnt 0).

#define T_STEPS  8192
#define SDIM     128
#define ODIM     64
#define NTHREADS 1024
#define NWAVES   (NTHREADS / 32)
#define LOG2PI   1.8378770664093453f

typedef __attribute__((ext_vector_type(2))) float v2f;
typedef __attribute__((ext_vector_type(8))) float v8f;

// Load a 2-element operand fragment for lane position rc at K-offset ka.
// kmajor=1: element (k, rc) at p[k*ld + rc]  (stride ld between k's)
// kmajor=0: element (rc, k) at p[rc*ld + k]  (contiguous in k)
__device__ __forceinline__ void frag_ld(const float* __restrict__ p, int ld, int kmajor,
                                        int rc, int ka, v2f& f)
{
    if (kmajor) { f.x = p[ka * ld + rc];  f.y = p[(ka + 1) * ld + rc]; }
    else        { f.x = p[rc * ld + ka];  f.y = p[rc * ld + ka + 1];   }
}

// D = op(A) x op(B), 16x16 tiles over 32 waves, K multiple of 4.
// transA: A[m][k] = Asrc[k*lda + m]; transB: B[k][n] = Bsrc[n*ldb + k].
__device__ __forceinline__ void wmma_mm(const float* __restrict__ Asrc, int lda, int transA,
                                        const float* __restrict__ Bsrc, int ldb, int transB,
                                        float* __restrict__ Dst, int ldd,
                                        int M, int N, int K, int tid)
{
    const int lane  = tid & 31;
    const int wave  = tid >> 5;
    const int lrow  = lane & 15;   // row/col within tile for this lane
    const int khalf = lane >> 4;   // 0: K=k,k+1   1: K=k+2,k+3  (16x4 A layout)
    const int akm   = transA ? 1 : 0;   // A k-major?
    const int bkm   = transB ? 0 : 1;   // B k-major?
    const int ntc   = N >> 4;
    const int ntiles = (M >> 4) * ntc;
    for (int t = wave; t < ntiles; t += NWAVES) {
        const int mt = (t / ntc) << 4;
        const int nt = (t % ntc) << 4;
        const int row = mt + lrow;
        const int col = nt + lrow;
        v8f c = {0.f, 0.f, 0.f, 0.f, 0.f, 0.f, 0.f, 0.f};
        v2f aC, bC, aN, bN;
        frag_ld(Asrc, lda, akm, row, 2 * khalf, aC);
        frag_ld(Bsrc, ldb, bkm, col, 2 * khalf, bC);
        for (int k = 0; k < K; k += 4) {
            const int kn = (k + 4 < K) ? (k + 4) : k;   // prefetch next stage
            frag_ld(Asrc, lda, akm, row, kn + 2 * khalf, aN);
            frag_ld(Bsrc, ldb, bkm, col, kn + 2 * khalf, bN);
            c = __builtin_amdgcn_wmma_f32_16x16x4_f32(false, aC, false, bC, (short)0, c,
                                                      false, false);
            aC = aN; bC = bN;
        }
#pragma unroll
        for (int i = 0; i < 8; ++i)
            Dst[(mt + i + 8 * khalf) * ldd + col] = c[i];
    }
}

__global__ __launch_bounds__(NTHREADS)
void KalmanFilter_59648505807192_kernel(const float* __restrict__ emis,
                                        const float* __restrict__ m0,
                                        const float* __restrict__ P0,
                                        const float* __restrict__ A,
                                        const float* __restrict__ Q,
                                        const float* __restrict__ H,
                                        const float* __restrict__ R,
                                        float* __restrict__ out)
{
    extern __shared__ float smem[];
    float* sA   = smem;                 // 128x128  A
    float* sH   = sA   + SDIM * SDIM;   // 64x128   H
    float* sP   = sH   + ODIM * SDIM;   // 128x128  P_filt (persists) / Pd scratch
    float* sPp  = sP   + SDIM * SDIM;   // 128x128  P_prior
    float* sT   = sPp  + SDIM * SDIM;   // 128x128  A@P ; later [HP | X]
    float* sS   = sT   + SDIM * SDIM;   // 64x64    S_inn -> L (in place)
    float* sm   = sS   + ODIM * ODIM;   // 128      m_prev / m_filt
    float* smp  = sm   + SDIM;          // 128      m_prior
    float* sy   = smp  + SDIM;          // 64       y_t
    float* sres = sy   + ODIM;          // 64       residual (kept)
    float* salp = sres + ODIM;          // 64       alpha (L^-1 residual)
    float* sred = salp + ODIM;          // 128      reduction scratch
    float* sll  = sred + SDIM;          // 1        running log-likelihood

    const int tid = threadIdx.x;

    for (int i = tid; i < SDIM * SDIM; i += NTHREADS) { sA[i] = A[i]; sP[i] = P0[i]; }
    for (int i = tid; i < ODIM * SDIM; i += NTHREADS) sH[i] = H[i];
    if (tid < SDIM) sm[tid] = m0[tid];
    if (tid == 0) sll[0] = 0.f;
    __syncthreads();

    float* out_means = out + 1;
    float* out_covs  = out + 1 + (size_t)T_STEPS * SDIM;

    for (int t = 0; t < T_STEPS; ++t) {
        // ---------------- predict ----------------
        if (tid < SDIM) {                       // m_prior = A @ m_prev
            float acc = 0.f;
            for (int k = 0; k < SDIM; ++k) acc += sA[tid * SDIM + k] * sm[k];
            smp[tid] = acc;
        }
        if (tid < ODIM) sy[tid] = emis[(size_t)t * ODIM + tid];
        if (tid == 0 && t + 1 < T_STEPS)
            __builtin_prefetch(&emis[(size_t)(t + 1) * ODIM], 0, 0);
        __syncthreads();

        wmma_mm(sA, SDIM, 0, sP, SDIM, 0, sT, SDIM, SDIM, SDIM, SDIM, tid);  // T = A P
        __syncthreads();
        wmma_mm(sT, SDIM, 0, sA, SDIM, 1, sPp, SDIM, SDIM, SDIM, SDIM, tid); // Pp = T A^T
        __syncthreads();
        for (int i = tid; i < SDIM * SDIM; i += NTHREADS) sPp[i] += Q[i];    // + Q
        if (tid < ODIM) {                       // residual = y - H m_prior
            float acc = 0.f;
            for (int k = 0; k < SDIM; ++k) acc += sH[tid * SDIM + k] * smp[k];
            sres[tid] = sy[tid] - acc;
        }
        __syncthreads();

        // ---------------- innovation ----------------
        float* sHP = sT;                 // 64x128
        float* sX  = sT + ODIM * SDIM;   // 64x128
        wmma_mm(sH, SDIM, 0, sPp, SDIM, 0, sHP, SDIM, ODIM, SDIM, SDIM, tid); // HP = H Pp
        __syncthreads();
        wmma_mm(sHP, SDIM, 0, sH, SDIM, 1, sS, ODIM, ODIM, ODIM, SDIM, tid);  // S = HP H^T
        __syncthreads();
        for (int i = tid; i < ODIM * ODIM; i += NTHREADS) sS[i] += R[i];      // + R
        for (int i = tid; i < ODIM * SDIM; i += NTHREADS) sX[i] = sHP[i];     // X <- HP
        if (tid < ODIM) salp[tid] = sres[tid];
        __syncthreads();

        // Cholesky S = L L^T, lower, in place (right-looking, column steps)
        for (int k = 0; k < ODIM; ++k) {
            if (tid == 0) sS[k * ODIM + k] = sqrtf(sS[k * ODIM + k]);
            __syncthreads();
            const float dinv = 1.0f / sS[k * ODIM + k];
            if (tid > k && tid < ODIM) sS[tid * ODIM + k] *= dinv;
            __syncthreads();
            for (int idx = tid; idx < ODIM * ODIM; idx += NTHREADS) {
                const int i = idx >> 6, j = idx & 63;
                if (j > k && j <= i)
                    sS[i * ODIM + j] -= sS[i * ODIM + k] * sS[j * ODIM + k];
            }
            __syncthreads();
        }

        // alpha = L^{-1} residual (column-oriented forward substitution)
        for (int j = 0; j < ODIM; ++j) {
            if (tid == 0) salp[j] /= sS[j * ODIM + j];
            __syncthreads();
            if (tid > j && tid < ODIM) salp[tid] -= sS[tid * ODIM + j] * salp[j];
            __syncthreads();
        }

        // log-likelihood contribution
        if (tid < ODIM) {
            sred[tid]        = logf(sS[tid * ODIM + tid]);
            sred[ODIM + tid] = salp[tid] * salp[tid];
        }
        __syncthreads();
        if (tid == 0) {
            float slog = 0.f, sa = 0.f;
            for (int i = 0; i < ODIM; ++i) { slog += sred[i]; sa += sred[ODIM + i]; }
            sll[0] += -0.5f * (float)ODIM * LOG2PI - slog - 0.5f * sa;
        }

        // X = S^{-1} HP : solve L Y = HP, then L^T X = Y, in place on sX
        for (int j = 0; j < ODIM; ++j) {
            const float dinv = 1.0f / sS[j * ODIM + j];
            if (tid < SDIM) sX[j * SDIM + tid] *= dinv;
            __syncthreads();
            for (int idx = tid; idx < (ODIM - 1 - j) * SDIM; idx += NTHREADS) {
                const int i = j + 1 + (idx >> 7);
                const int c = idx & (SDIM - 1);
                sX[i * SDIM + c] -= sS[i * ODIM + j] * sX[j * SDIM + c];
            }
            __syncthreads();
        }
        for (int j = ODIM - 1; j >= 0; --j) {
            const float dinv = 1.0f / sS[j * ODIM + j];
            if (tid < SDIM) sX[j * SDIM + tid] *= dinv;
            __syncthreads();
            for (int idx = tid; idx < j * SDIM; idx += NTHREADS) {
                const int i = idx >> 7;
                const int c = idx & (SDIM - 1);
                sX[i * SDIM + c] -= sS[j * ODIM + i] * sX[j * SDIM + c];  // L^T[i][j]=L[j][i]
            }
            __syncthreads();
        }

        // ---------------- update ----------------
        if (tid < SDIM) {                       // m_filt = m_prior + X^T residual
            float acc = smp[tid];
            for (int o = 0; o < ODIM; ++o) acc += sX[o * SDIM + tid] * sres[o];
            sm[tid] = acc;
        }
        __syncthreads();
        // Pd = X^T @ HP  (128x128, K=64) into sP (old P dead)
        wmma_mm(sX, SDIM, 1, sHP, SDIM, 0, sP, SDIM, SDIM, SDIM, ODIM, tid);
        __syncthreads();
        float* covt = out_covs + (size_t)t * SDIM * SDIM;
        for (int i = tid; i < SDIM * SDIM; i += NTHREADS) {
            const float v = sPp[i] - sP[i];     // P_filt = Pp - X^T HP
            sP[i]  = v;
            covt[i] = v;
        }
        if (tid < SDIM) out_means[(size_t)t * SDIM + tid] = sm[tid];
        __syncthreads();
    }

    if (tid == 0) out[0] = sll[0];
}

extern "C" void kernel_launch(void* const* d_in, const int* in_sizes, int n_in,
                              void* d_out, int out_size, void* d_ws, size_t ws_size,
                              hipStream_t stream) {
    (void)in_sizes; (void)n_in; (void)d_ws; (void)ws_size; (void)out_size;
    const float* emis = (const float*)d_in[0];
    const float* m0   = (const float*)d_in[1];
    const float* P0   = (const float*)d_in[2];
    const float* A    = (const float*)d_in[3];
    const float* Q    = (const float*)d_in[4];
    const float* H    = (const float*)d_in[5];
    const float* R    = (const float*)d_in[6];
    float* out = (float*)d_out;

    const size_t smem_bytes =
        (size_t)(SDIM * SDIM      /* sA  */
               + ODIM * SDIM      /* sH  */
               + SDIM * SDIM      /* sP  */
               + SDIM * SDIM      /* sPp */
               + SDIM * SDIM      /* sT  */
               + ODIM * ODIM      /* sS  */
               + SDIM + SDIM + ODIM + ODIM + ODIM + SDIM + 8) * sizeof(float);

    KalmanFilter_59648505807192_kernel<<<dim3(1), dim3(NTHREADS), smem_bytes, stream>>>(
        emis, m0, P0, A, Q, H, R, out);
}